// ClusterisedLinearNetwork_72713796321290
// MI455X (gfx1250) — compile-verified
//
#include <hip/hip_runtime.h>
#include <math.h>

// ---------------------------------------------------------------------------
// ClusterisedLinearNetwork on MI455X (gfx1250)
//
// Per point p: enc = posenc(X[p]) in registers (double-angle recurrence),
// out[p, 0..2] = dot(enc, W[3*cid+j]) with W staged in LDS (153.6 KB fits
// the 320 KB WGP LDS) via CDNA5 async global->LDS copies (ASYNCcnt path).
// WMMA intentionally unused: per-point gathered weights have no shared
// B operand, and the dense-GEMM formulation does 128x excess FLOPs.
// ---------------------------------------------------------------------------

#define NUM_FREQS 10
#define NCHAN     5
#define DIMS      (NCHAN * 2 * NUM_FREQS)   // 100 floats per point
#define ROW_VEC4  (DIMS / 4)                // 25 float4 per W row
#define WROWS     384                       // 3 * 128 clusters
#define W_VEC4    (WROWS * ROW_VEC4)        // 9600 float4 = 153600 bytes
#define BLOCK     256
#define GRID      512

#define AS1 __attribute__((address_space(1)))
#define AS3 __attribute__((address_space(3)))

typedef int v4i __attribute__((ext_vector_type(4)));

__device__ __forceinline__ void async_wait_all() {
#if __has_builtin(__builtin_amdgcn_s_wait_asynccnt)
    __builtin_amdgcn_s_wait_asynccnt(0);
#else
    asm volatile("s_wait_asynccnt 0" ::: "memory");
#endif
}

__global__ __launch_bounds__(BLOCK)
void clusterised_linear_kernel(const float* __restrict__ X,
                               const int*   __restrict__ cids,
                               const float* __restrict__ W,
                               float*       __restrict__ out,
                               int n) {
    __shared__ float4 sW[W_VEC4];   // 153600 B static LDS (WGP has 320 KB)

    // ---- Stage all of W into LDS with async global->LDS B128 copies ----
    {
        const float4* __restrict__ W4 = (const float4*)W;
        for (int i = threadIdx.x; i < W_VEC4; i += BLOCK) {
#if __has_builtin(__builtin_amdgcn_global_load_async_to_lds_b128)
            __builtin_amdgcn_global_load_async_to_lds_b128(
                (AS1 v4i*)(W4 + i),          // global src (addrspace 1)
                (AS3 v4i*)(sW + i),          // LDS dst    (addrspace 3)
                /*offset=*/0, /*cpol=*/0);
#else
            sW[i] = W4[i];          // fallback: plain LDS store path
#endif
        }
        async_wait_all();           // this wave's async loads landed in LDS
    }
    __syncthreads();                // all waves' fills visible

    const int tid0   = (int)(blockIdx.x * BLOCK + threadIdx.x);
    const int stride = GRID * BLOCK;

    for (int p = tid0; p < n; p += stride) {
        // ---- load inputs (coalesced: 20 B + 4 B per lane) ----
        float xs[NCHAN];
#pragma unroll
        for (int c = 0; c < NCHAN; ++c) xs[c] = X[p * NCHAN + c];
        const int cl = cids[p];

        // ---- positional encoding via double-angle recurrence ----
        // enc4[c*5 + j] = { sin(2^(2j) x_c), cos(2^(2j) x_c),
        //                   sin(2^(2j+1) x_c), cos(2^(2j+1) x_c) }
        // which matches the reference reshape layout exactly.
        float4 enc4[ROW_VEC4];
#pragma unroll
        for (int c = 0; c < NCHAN; ++c) {
            float s, cv;
            sincosf(xs[c], &s, &cv);            // one precise sincos per chan
#pragma unroll
            for (int j = 0; j < NUM_FREQS / 2; ++j) {
                const float s2 = 2.0f * s * cv;              // sin(2a)
                const float c2 = fmaf(cv + cv, cv, -1.0f);   // cos(2a)
                enc4[c * 5 + j] = make_float4(s, cv, s2, c2);
                s  = 2.0f * s2 * c2;                         // sin(4a)
                cv = fmaf(c2 + c2, c2, -1.0f);               // cos(4a)
            }
        }

        // ---- 3 gathered dot products against LDS-resident W rows ----
        const float4* __restrict__ r0 = &sW[(3 * cl) * ROW_VEC4];
        float a0 = 0.0f, a1 = 0.0f, a2 = 0.0f;
#pragma unroll
        for (int t = 0; t < ROW_VEC4; ++t) {
            const float4 e  = enc4[t];
            const float4 w0 = r0[t];
            const float4 w1 = r0[t + ROW_VEC4];
            const float4 w2 = r0[t + 2 * ROW_VEC4];
            a0 = fmaf(e.x, w0.x, fmaf(e.y, w0.y, fmaf(e.z, w0.z, fmaf(e.w, w0.w, a0))));
            a1 = fmaf(e.x, w1.x, fmaf(e.y, w1.y, fmaf(e.z, w1.z, fmaf(e.w, w1.w, a1))));
            a2 = fmaf(e.x, w2.x, fmaf(e.y, w2.y, fmaf(e.z, w2.z, fmaf(e.w, w2.w, a2))));
        }

        // ---- coalesced 12 B/lane store ----
        out[p * 3 + 0] = a0;
        out[p * 3 + 1] = a1;
        out[p * 3 + 2] = a2;
    }
}

extern "C" void kernel_launch(void* const* d_in, const int* in_sizes, int n_in,
                              void* d_out, int out_size, void* d_ws, size_t ws_size,
                              hipStream_t stream) {
    const float* X    = (const float*)d_in[0];   // [N, 5] fp32
    const int*   cids = (const int*)  d_in[1];   // [N] int32
    const float* W    = (const float*)d_in[2];   // [384, 100] fp32
    float*       out  = (float*)d_out;           // [N, 3] fp32
    const int n = in_sizes[1];                   // N = number of points

    clusterised_linear_kernel<<<GRID, BLOCK, 0, stream>>>(X, cids, W, out, n);
}